// DistillLoss_85237920957114
// MI455X (gfx1250) — compile-verified
//
#include <hip/hip_runtime.h>

// Problem constants (from the reference)
#define DIMD   65536      // feature dim per row
#define NROWS  1024       // rows per matrix (B2); also the number of (q,v) pairs
#define BHALF  512        // B2/2
#define TPB    256        // threads per block (8 wave32)
#define NBLK   240        // persistent blocks: bounds concurrent L2 working set
                          // to 240 * 512KB = 120 MB < 192 MB global L2

typedef __attribute__((ext_vector_type(2))) float v2f;
typedef __attribute__((ext_vector_type(4))) float v4f;
typedef __attribute__((ext_vector_type(8))) float v8f;

#define FLT_BIG 3.402823466e+38f

// teacher temperature schedule: linspace(0.07, 0.04, 30) then 0.04
__device__ __forceinline__ float teacher_invT(const int* __restrict__ epoch_ptr) {
    int e = *epoch_ptr;                          // uniform scalar load
    float temp = (e >= 30) ? 0.04f
                           : (0.07f + (0.04f - 0.07f) * ((float)e * (1.0f / 29.0f)));
    return __builtin_amdgcn_rcpf(temp);
}

// online-softmax combine: (m,s) <- merge (m,s),(om,os); sentinel is -FLT_BIG (never -inf)
__device__ __forceinline__ void combine_ms(float& m, float& s, float om, float os) {
    float nm = fmaxf(m, om);
    s = s * __expf(m - nm) + os * __expf(om - nm);
    m = nm;
}

// ---------------------------------------------------------------------------
// Fused per-pair kernel (persistent, grid-stride over pairs).
// Pair p: q-row = teacher row p (temp from schedule),
//         v-row = student row (p<512 ? p+512 : p-512) (temp 0.1).
// Phase 1: stream both rows once from HBM (RT policy: lines stay in L2),
//          online (max, sum-exp) for each row.
// Phase 2: re-stream both rows — L2 hits — with NT (last-use) policy so the
//          consumed lines drain from L2 early, then accumulate
//          Sqv = sum q^2/v  and  Sneg = sum (1-q)^2/(1-v).
// ---------------------------------------------------------------------------
__global__ __launch_bounds__(TPB)
void DistillLoss_pair_kernel(const float* __restrict__ student,
                             const float* __restrict__ teacher,
                             const int*   __restrict__ epoch,
                             float* __restrict__ Sqv,
                             float* __restrict__ Sneg) {
    const float invTt = teacher_invT(epoch);
    const int t    = threadIdx.x;
    const int wave = t >> 5, lane = t & 31;

    __shared__ float red0[TPB / 32];   // per-wave partials, reused across stages
    __shared__ float red1[TPB / 32];
    __shared__ float red2[TPB / 32];
    __shared__ float red3[TPB / 32];
    __shared__ float bc[4];            // broadcast: tmax, rcp(tsum), smax, rcp(ssum)

    for (int p = blockIdx.x; p < NROWS; p += gridDim.x) {
        const int srow = (p < BHALF) ? (p + BHALF) : (p - BHALF);
        const v4f* t4 = reinterpret_cast<const v4f*>(teacher + (size_t)p    * DIMD);
        const v4f* s4 = reinterpret_cast<const v4f*>(student + (size_t)srow * DIMD);

        // ---- Phase 1: online softmax normalizers for both rows (RT loads) --
        float mt = -FLT_BIG, st = 0.0f;     // teacher row (scaled by invTt)
        float ms = -FLT_BIG, ss = 0.0f;     // student row (scaled by 10)
        for (int i = 0; i < DIMD / 4 / TPB; ++i) {
            const int idx = i * TPB + t;
            __builtin_prefetch(t4 + idx + 8 * TPB, 0, 0);   // global_prefetch_b8
            __builtin_prefetch(s4 + idx + 8 * TPB, 0, 0);
            v4f tv = t4[idx];
            v4f sv = s4[idx];
            #pragma unroll
            for (int k = 0; k < 4; ++k) {
                float xt = tv[k] * invTt;
                if (xt > mt) { st = st * __expf(mt - xt) + 1.0f; mt = xt; }
                else         { st += __expf(xt - mt); }
                float xs = sv[k] * 10.0f;
                if (xs > ms) { ss = ss * __expf(ms - xs) + 1.0f; ms = xs; }
                else         { ss += __expf(xs - ms); }
            }
        }
        #pragma unroll
        for (int off = 16; off > 0; off >>= 1) {
            combine_ms(mt, st, __shfl_xor(mt, off, 32), __shfl_xor(st, off, 32));
            combine_ms(ms, ss, __shfl_xor(ms, off, 32), __shfl_xor(ss, off, 32));
        }
        if (lane == 0) { red0[wave] = mt; red1[wave] = st; red2[wave] = ms; red3[wave] = ss; }
        __syncthreads();
        if (t == 0) {
            float Mt = red0[0], St = red1[0], Ms = red2[0], Ss = red3[0];
            #pragma unroll
            for (int w = 1; w < TPB / 32; ++w) {
                combine_ms(Mt, St, red0[w], red1[w]);
                combine_ms(Ms, Ss, red2[w], red3[w]);
            }
            bc[0] = Mt; bc[1] = __builtin_amdgcn_rcpf(St);
            bc[2] = Ms; bc[3] = __builtin_amdgcn_rcpf(Ss);
        }
        __syncthreads();
        const float tmax = bc[0], rt = bc[1];
        const float smax = bc[2], rs = bc[3];

        // ---- Phase 2: Renyi sums (L2-resident re-read; NT = last use) ------
        float acc1 = 0.0f, acc2 = 0.0f;
        for (int i = 0; i < DIMD / 4 / TPB; ++i) {
            const int idx = i * TPB + t;
            v4f tv = __builtin_nontemporal_load(t4 + idx);
            v4f sv = __builtin_nontemporal_load(s4 + idx);
            #pragma unroll
            for (int k = 0; k < 4; ++k) {
                float q = __expf(tv[k] * invTt - tmax) * rt;   // teacher softmax elem
                float v = __expf(sv[k] * 10.0f - smax) * rs;   // student softmax elem
                acc1 += q * q * __builtin_amdgcn_rcpf(v);
                float iq = 1.0f - q, iv = 1.0f - v;
                acc2 += iq * iq * __builtin_amdgcn_rcpf(iv);
            }
        }
        #pragma unroll
        for (int off = 16; off > 0; off >>= 1) {
            acc1 += __shfl_xor(acc1, off, 32);
            acc2 += __shfl_xor(acc2, off, 32);
        }
        __syncthreads();                 // red* free for reuse
        if (lane == 0) { red0[wave] = acc1; red1[wave] = acc2; }
        __syncthreads();
        if (t == 0) {
            float r1 = red0[0], r2 = red1[0];
            #pragma unroll
            for (int w = 1; w < TPB / 32; ++w) { r1 += red0[w]; r2 += red1[w]; }
            Sqv[p]  = r1;
            Sneg[p] = r2;
        }
        __syncthreads();                 // protect shared state before next pair
    }
}

// ---------------------------------------------------------------------------
// Finalize: single wave32. loss = (1/1024) * sum_p [ log Sqv_p + BETA*log Sneg_p ].
// 32 lane partials reduced with v_wmma_f32_16x16x4_f32 (A = ones, B = data:
// D[m][n] = column sums of B; sums are order-independent), then 4 shuffle-adds.
// EXEC is all ones (exactly 32 threads, no divergence) as WMMA requires.
// ---------------------------------------------------------------------------
__global__ __launch_bounds__(32)
void DistillLoss_finalize_kernel(const float* __restrict__ Sqv,
                                 const float* __restrict__ Sneg,
                                 float* __restrict__ out) {
    const int lane = threadIdx.x;
    float acc = 0.0f;
    #pragma unroll 4
    for (int i = 0; i < NROWS / 32; ++i) {
        const int p = i * 32 + lane;
        acc += __logf(Sqv[p]) + 1.0f /*BETA*/ * __logf(Sneg[p]);
    }

    v2f a; a.x = 1.0f; a.y = 1.0f;     // A = 16x4 ones
    v2f b; b.x = acc;  b.y = 0.0f;     // B rows K0/K2 carry the 32 lane partials
    v8f c = {};
    c = __builtin_amdgcn_wmma_f32_16x16x4_f32(
            /*neg_a=*/false, a, /*neg_b=*/false, b,
            /*c_mod=*/(short)0, c, /*reuse_a=*/false, /*reuse_b=*/false);
    // c[0] on lane n (n<16): D[0][n] = partial_n + partial_{n+16}; lanes 16..31 identical
    float colsum = c[0];
    #pragma unroll
    for (int off = 8; off > 0; off >>= 1) colsum += __shfl_xor(colsum, off, 32);

    if (lane == 0) out[0] = colsum * (1.0f / 1024.0f);
}

// ---------------------------------------------------------------------------
extern "C" void kernel_launch(void* const* d_in, const int* in_sizes, int n_in,
                              void* d_out, int out_size, void* d_ws, size_t ws_size,
                              hipStream_t stream) {
    const float* student = (const float*)d_in[0];   // (1024, 65536) f32
    const float* teacher = (const float*)d_in[1];   // (1024, 65536) f32
    const int*   epoch   = (const int*)d_in[2];     // scalar

    float* ws   = (float*)d_ws;
    float* Sqv  = ws;                   // 1024 floats
    float* Sneg = ws + NROWS;           // 1024 floats  (8 KB total)

    DistillLoss_pair_kernel<<<NBLK, TPB, 0, stream>>>(
        student, teacher, epoch, Sqv, Sneg);

    DistillLoss_finalize_kernel<<<1, 32, 0, stream>>>(Sqv, Sneg, (float*)d_out);
}